// TanhAttention_89781996356126
// MI455X (gfx1250) — compile-verified
//
#include <hip/hip_runtime.h>
#include <math.h>

// ---------------------------------------------------------------------------
// TanhAttention for MI455X (gfx1250, wave32, WMMA).
//   B=4, T_in=256, T_mem=256, D=512.
//   Stage 1: in_proj = inputs@W_in + b_in ; mem_projT = (memory@W_mem)^T
//            -> V_WMMA_F32_16X16X4_F32, one wave per 32x32 tile (2x2 frags).
//   Stage 2: score/tanh/mask/softmax fused, 4 input rows per block, LDS-staged.
//   Stage 3: out = weight@memory -> V_WMMA_F32_16X16X4_F32, 32x32 tiles.
// ---------------------------------------------------------------------------

typedef __attribute__((ext_vector_type(2))) float v2f;
typedef __attribute__((ext_vector_type(8))) float v8f;

#define BB    4
#define T_IN  256
#define T_MEM 256
#define DD    512
#define NEGV  (-1e30f)
#define IBLK  4

static __device__ __forceinline__ v8f wmma_f32(v2f a, v2f b, v8f c) {
    return __builtin_amdgcn_wmma_f32_16x16x4_f32(
        /*neg_a=*/false, a, /*neg_b=*/false, b,
        /*c_mod=*/(short)0, c, /*reuse_a=*/false, /*reuse_b=*/false);
}

// ---------------------------------------------------------------------------
// Kernel 1: dual projection GEMM, 32x32 register tile per wave.
//   sel==0: in_proj[m,n]     = sum_k inputs[m,k]*W_in[k,n] + b_in[n]
//   sel==1: mem_projT[b,n,j] = sum_k memory[b*256+j,k]*W_mem[k,n]  (transposed)
// A frag (16x4 f32): lanes 0-15 hold M rows with K=0,1 in v0,v1; lanes 16-31
// hold the same rows with K=2,3. B frag (4x16): v0 = K=0 row (lanes 0-15) /
// K=2 row (lanes 16-31); v1 = K=1 / K=3. C/D: vgpr r -> (M=r, N=lane) for
// lanes 0-15, (M=r+8, N=lane-16) for lanes 16-31.
// ---------------------------------------------------------------------------
static __device__ __forceinline__ void store_proj_tile(
    v8f acc, int mbase, int col, int half, int sel, const float* __restrict__ b_in,
    float* __restrict__ in_proj, float* __restrict__ mem_projT) {
#pragma unroll
    for (int r = 0; r < 8; ++r) {
        const int m = mbase + r + 8 * half;      // global row 0..1023
        const float v = acc[r];
        if (sel == 0) {
            in_proj[(size_t)m * DD + col] = v + b_in[col];
        } else {
            const int b = m >> 8;                // batch
            const int j = m & 255;               // memory position
            mem_projT[((size_t)b * DD + col) * T_MEM + j] = v;
        }
    }
}

__global__ __launch_bounds__(32)
void tanhattn_proj_gemm(const float* __restrict__ inputs,
                        const float* __restrict__ memory,
                        const float* __restrict__ W_in,
                        const float* __restrict__ b_in,
                        const float* __restrict__ W_mem,
                        float* __restrict__ in_proj,    // [1024, 512]
                        float* __restrict__ mem_projT)  // [B, 512, 256]
{
    const int sel  = blockIdx.z;                 // 0 -> inputs, 1 -> memory
    const float* A = sel ? memory : inputs;      // [1024, 512] row-major
    const float* W = sel ? W_mem  : W_in;        // [512, 512]  row-major

    const int tm   = blockIdx.x;                 // 0..31  (M/32 tiles)
    const int tn   = blockIdx.y;                 // 0..15  (N/32 tiles)
    const int lane = threadIdx.x;
    const int half = lane >> 4;
    const int lr   = lane & 15;

    const float* arow0 = A + (size_t)(tm * 32 + lr) * DD;
    const float* arow1 = arow0 + (size_t)16 * DD;
    const int    col0  = tn * 32 + lr;
    const int    col1  = col0 + 16;

    v8f acc00 = {}, acc01 = {}, acc10 = {}, acc11 = {};
    for (int kk = 0; kk < DD; kk += 4) {
        const int k0 = kk + 2 * half;
        __builtin_prefetch(arow0 + k0 + 64, 0, 1);   // global_prefetch_b8

        v2f a0, a1, b0, b1;
        a0.x = arow0[k0];
        a0.y = arow0[k0 + 1];
        a1.x = arow1[k0];
        a1.y = arow1[k0 + 1];
        const float* wr0 = W + (size_t)k0 * DD;
        const float* wr1 = wr0 + DD;
        b0.x = wr0[col0];
        b0.y = wr1[col0];
        b1.x = wr0[col1];
        b1.y = wr1[col1];

        acc00 = wmma_f32(a0, b0, acc00);
        acc01 = wmma_f32(a0, b1, acc01);
        acc10 = wmma_f32(a1, b0, acc10);
        acc11 = wmma_f32(a1, b1, acc11);
    }

    const int m0 = tm * 32;
    store_proj_tile(acc00, m0,      col0, half, sel, b_in, in_proj, mem_projT);
    store_proj_tile(acc01, m0,      col1, half, sel, b_in, in_proj, mem_projT);
    store_proj_tile(acc10, m0 + 16, col0, half, sel, b_in, in_proj, mem_projT);
    store_proj_tile(acc11, m0 + 16, col1, half, sel, b_in, in_proj, mem_projT);
}

// ---------------------------------------------------------------------------
// Kernel 2: fused score + tanh + mask + softmax.
// One block per (i-block of 4, b): 256 threads, thread t owns column j = t and
// computes IBLK=4 rows. Each mem_projT load feeds 4 tanh/fma pairs (4x traffic
// amortization); 4 independent FMA chains hide TRANS latency.
// ---------------------------------------------------------------------------
__global__ __launch_bounds__(256)
void tanhattn_score_softmax(const float* __restrict__ in_proj,   // [1024,512]
                            const float* __restrict__ mem_projT, // [B,512,256]
                            const float* __restrict__ w_final,   // [512]
                            const int*   __restrict__ in_len,    // [B]
                            const int*   __restrict__ mem_len,   // [B]
                            float* __restrict__ weight)          // [B,256,256]
{
    __shared__ float s_in[IBLK][DD];
    __shared__ float s_w[DD];
    __shared__ float s_red[256];

    const int i0 = blockIdx.x * IBLK;  // first input position of this block
    const int b  = blockIdx.y;         // batch
    const int t  = threadIdx.x;        // memory position j

    for (int d = t; d < DD; d += 256) {
        s_w[d] = w_final[d];
#pragma unroll
        for (int ii = 0; ii < IBLK; ++ii)
            s_in[ii][d] = in_proj[((size_t)b * T_IN + i0 + ii) * DD + d];
    }
    __syncthreads();

    const float* mp = mem_projT + (size_t)b * DD * T_MEM + t;  // column t
    float acc[IBLK] = {0.0f, 0.0f, 0.0f, 0.0f};
#pragma unroll 2
    for (int d = 0; d < DD; ++d) {
        const float m = mp[(size_t)d * T_MEM];   // coalesced across threads
        const float w = s_w[d];
#pragma unroll
        for (int ii = 0; ii < IBLK; ++ii)
            acc[ii] = fmaf(w, tanhf(s_in[ii][d] + m), acc[ii]);
    }

    const int ilen = in_len[b];
    const int mlen = mem_len[b];

    for (int ii = 0; ii < IBLK; ++ii) {
        float score = acc[ii];
        if (i0 + ii >= ilen) score = NEGV;   // row mask
        if (t >= mlen)       score = NEGV;   // column mask

        // --- max reduce over 256 threads ---
        s_red[t] = score;
        __syncthreads();
        for (int s = 128; s > 0; s >>= 1) {
            if (t < s) s_red[t] = fmaxf(s_red[t], s_red[t + s]);
            __syncthreads();
        }
        const float mx = s_red[0];
        __syncthreads();

        const float e = __expf(score - mx);

        // --- sum reduce ---
        s_red[t] = e;
        __syncthreads();
        for (int s = 128; s > 0; s >>= 1) {
            if (t < s) s_red[t] = s_red[t] + s_red[t + s];
            __syncthreads();
        }
        const float sum = s_red[0];
        __syncthreads();   // s_red reused next iteration

        weight[((size_t)b * T_IN + i0 + ii) * T_MEM + t] = e / sum;
    }
}

// ---------------------------------------------------------------------------
// Kernel 3: out[b] = weight[b] (256x256) @ memory[b] (256x512), f32 WMMA,
// 32x32 register tile per wave.
// ---------------------------------------------------------------------------
__global__ __launch_bounds__(32)
void tanhattn_out_gemm(const float* __restrict__ weight,  // [B,256,256]
                       const float* __restrict__ memory,  // [B,256,512]
                       float* __restrict__ out)           // [B,256,512]
{
    const int b    = blockIdx.z;
    const int tm   = blockIdx.x;                 // 0..7   (256/32)
    const int tn   = blockIdx.y;                 // 0..15  (512/32)
    const int lane = threadIdx.x;
    const int half = lane >> 4;
    const int lr   = lane & 15;

    const float* Wt = weight + (size_t)b * T_IN * T_MEM;
    const float* Mm = memory + (size_t)b * T_MEM * DD;

    const float* arow0 = Wt + (size_t)(tm * 32 + lr) * T_MEM;
    const float* arow1 = arow0 + (size_t)16 * T_MEM;
    const int    col0  = tn * 32 + lr;
    const int    col1  = col0 + 16;

    v8f acc00 = {}, acc01 = {}, acc10 = {}, acc11 = {};
    for (int kk = 0; kk < T_MEM; kk += 4) {
        const int k0 = kk + 2 * half;
        __builtin_prefetch(arow0 + k0 + 64, 0, 1);

        v2f a0, a1, b0, b1;
        a0.x = arow0[k0];
        a0.y = arow0[k0 + 1];
        a1.x = arow1[k0];
        a1.y = arow1[k0 + 1];
        const float* mr0 = Mm + (size_t)k0 * DD;
        const float* mr1 = mr0 + DD;
        b0.x = mr0[col0];
        b0.y = mr1[col0];
        b1.x = mr0[col1];
        b1.y = mr1[col1];

        acc00 = wmma_f32(a0, b0, acc00);
        acc01 = wmma_f32(a0, b1, acc01);
        acc10 = wmma_f32(a1, b0, acc10);
        acc11 = wmma_f32(a1, b1, acc11);
    }

    float* orow = out + (size_t)b * T_IN * DD;
#pragma unroll
    for (int r = 0; r < 8; ++r) {
        const int m0 = tm * 32 + r + 8 * half;
        orow[(size_t)m0 * DD + col0]        = acc00[r];
        orow[(size_t)m0 * DD + col1]        = acc01[r];
        orow[(size_t)(m0 + 16) * DD + col0] = acc10[r];
        orow[(size_t)(m0 + 16) * DD + col1] = acc11[r];
    }
}

// ---------------------------------------------------------------------------
// Host-side launcher.
// Inputs (setup_inputs order):
//   0 inputs [B,256,512] f32      1 memory [B,256,512] f32
//   2 input_lengths [B] i32       3 memory_lengths [B] i32
//   4 W_in [512,512] f32          5 b_in [512] f32
//   6 W_mem [512,512] f32         7 w_final [512] f32
// Output: [B,256,512] f32.
// Workspace (floats): in_proj 1024*512 | mem_projT 4*512*256 | weight 4*256*256
// ---------------------------------------------------------------------------
extern "C" void kernel_launch(void* const* d_in, const int* in_sizes, int n_in,
                              void* d_out, int out_size, void* d_ws, size_t ws_size,
                              hipStream_t stream) {
    const float* inputs  = (const float*)d_in[0];
    const float* memory  = (const float*)d_in[1];
    const int*   in_len  = (const int*)d_in[2];
    const int*   mem_len = (const int*)d_in[3];
    const float* W_in    = (const float*)d_in[4];
    const float* b_in    = (const float*)d_in[5];
    const float* W_mem   = (const float*)d_in[6];
    const float* w_final = (const float*)d_in[7];
    float*       out     = (float*)d_out;

    float* ws        = (float*)d_ws;
    float* in_proj   = ws;                                   // 1024*512
    float* mem_projT = ws + (size_t)1024 * 512;              // 4*512*256
    float* weight    = mem_projT + (size_t)BB * DD * T_MEM;  // 4*256*256

    // Stage 1: both projections (z selects which one). 32x32 tiles.
    {
        dim3 grid(1024 / 32, DD / 32, 2);
        dim3 block(32, 1, 1);
        tanhattn_proj_gemm<<<grid, block, 0, stream>>>(
            inputs, memory, W_in, b_in, W_mem, in_proj, mem_projT);
    }
    // Stage 2: score + tanh + mask + softmax (4 input rows per block).
    {
        dim3 grid(T_IN / IBLK, BB, 1);
        dim3 block(256, 1, 1);
        tanhattn_score_softmax<<<grid, block, 0, stream>>>(
            in_proj, mem_projT, w_final, in_len, mem_len, weight);
    }
    // Stage 3: weighted sum of memory. 32x32 tiles.
    {
        dim3 grid(T_IN / 32, DD / 32, BB);
        dim3 block(32, 1, 1);
        tanhattn_out_gemm<<<grid, block, 0, stream>>>(weight, memory, out);
    }
}